// E88FLAHybrid_57939108823757
// MI455X (gfx1250) — compile-verified
//
#include <hip/hip_runtime.h>
#include <math.h>

// Problem constants (from reference)
#define DIM 1024
#define H_  8
#define N_  32
#define DC_ 4
#define KD_ 256
#define VD_ 512
#define HV_ 64
#define B_  2
#define T_  2048
#define MROWS (B_ * T_)   // 4096

typedef float v2f __attribute__((ext_vector_type(2)));
typedef float v4f __attribute__((ext_vector_type(4)));
typedef float v8f __attribute__((ext_vector_type(8)));

__device__ __forceinline__ float fast_sigmoid(float x) {
    return 1.0f / (1.0f + __expf(-x));
}
__device__ __forceinline__ float fast_tanh(float x) {
    // tanh(x) = 1 - 2/(exp(2x)+1); saturates correctly for |x| large
    float e = __expf(2.0f * x);
    return 1.0f - 2.0f / (e + 1.0f);
}
__device__ __forceinline__ float fast_silu(float x) { return x * fast_sigmoid(x); }

// ---------------------------------------------------------------------------
// WMMA GEMM (f32, 16x16x4):  Y[M,Nc] = X[M,K] * W[Nc,K]^T
// One wave per block; each wave computes a 16(M) x 128(N) tile (8 accumulators).
// K is consumed in chunks of 8 via b128 loads: lane half=0 loads k0..k0+3,
// half=1 loads k0+4..k0+7. Two WMMAs per chunk per acc: elements {0,1} cover
// K-slots {k0,k0+1,k0+4,k0+5}; elements {2,3} cover {k0+2,k0+3,k0+6,k0+7}.
// (Valid because A and B use identical slot->k assignment; sum over K is
// permutation invariant.) Next chunk is prefetched before the current chunk's
// WMMAs so loads overlap compute instead of s_wait_loadcnt 0 per load.
// ---------------------------------------------------------------------------
__device__ __forceinline__ void wmma_chunk8(v4f a, const v4f* b, v8f* acc) {
    const v2f a01 = __builtin_shufflevector(a, a, 0, 1);
    const v2f a23 = __builtin_shufflevector(a, a, 2, 3);
#pragma unroll
    for (int j = 0; j < 8; ++j) {
        const v2f b01 = __builtin_shufflevector(b[j], b[j], 0, 1);
        acc[j] = __builtin_amdgcn_wmma_f32_16x16x4_f32(false, a01, false, b01,
                                                       (short)0, acc[j], false, false);
    }
#pragma unroll
    for (int j = 0; j < 8; ++j) {
        const v2f b23 = __builtin_shufflevector(b[j], b[j], 2, 3);
        acc[j] = __builtin_amdgcn_wmma_f32_16x16x4_f32(false, a23, false, b23,
                                                       (short)0, acc[j], false, false);
    }
}

__global__ __launch_bounds__(32)
void gemm_xwT_wmma(const float* __restrict__ X, const float* __restrict__ W,
                   float* __restrict__ Y, int K, int Nc) {
    const int n0   = blockIdx.x * 128;
    const int m0   = blockIdx.y * 16;
    const int lane = threadIdx.x;
    const int half = lane >> 4;     // 0: k0..k0+3, 1: k0+4..k0+7
    const int lr   = lane & 15;

    const float* xrow = X + (size_t)(m0 + lr) * K + (half << 2);
    const float* wr[8];
#pragma unroll
    for (int j = 0; j < 8; ++j)
        wr[j] = W + (size_t)(n0 + 16 * j + lr) * K + (half << 2);

    v8f acc[8];
#pragma unroll
    for (int j = 0; j < 8; ++j) acc[j] = (v8f){};

    // prologue: load chunk 0
    v4f a_c = *(const v4f*)(xrow);
    v4f b_c[8];
#pragma unroll
    for (int j = 0; j < 8; ++j) b_c[j] = *(const v4f*)(wr[j]);

    for (int k0 = 0; k0 < K - 8; k0 += 8) {
        // prefetch chunk k0+8
        v4f a_n = *(const v4f*)(xrow + k0 + 8);
        v4f b_n[8];
#pragma unroll
        for (int j = 0; j < 8; ++j) b_n[j] = *(const v4f*)(wr[j] + k0 + 8);

        wmma_chunk8(a_c, b_c, acc);   // consume chunk k0 while loads in flight

        a_c = a_n;
#pragma unroll
        for (int j = 0; j < 8; ++j) b_c[j] = b_n[j];
    }
    wmma_chunk8(a_c, b_c, acc);       // epilogue: chunk K-8

#pragma unroll
    for (int r = 0; r < 8; ++r) {
        const int m = m0 + r + half * 8;
        float* yrow = Y + (size_t)m * Nc + n0 + lr;
#pragma unroll
        for (int j = 0; j < 8; ++j) yrow[16 * j] = acc[j][r];
    }
}

// ---------------------------------------------------------------------------
// Wa projection (Nout = 8, too narrow for a WMMA tile): one wave per (m,h).
// ---------------------------------------------------------------------------
__global__ __launch_bounds__(256)
void a_proj_kernel(const float* __restrict__ x, const float* __restrict__ Wa,
                   float* __restrict__ ap) {
    const int wave_g = blockIdx.x * 8 + (threadIdx.x >> 5);
    const int lane   = threadIdx.x & 31;
    const int m = wave_g >> 3;   // / H_
    const int h = wave_g & 7;    // % H_
    const float* xr = x  + (size_t)m * DIM;
    const float* wr = Wa + (size_t)h * DIM;
    float s = 0.0f;
    for (int k = lane; k < DIM; k += 32) s = fmaf(xr[k], wr[k], s);
#pragma unroll
    for (int msk = 16; msk >= 1; msk >>= 1) s += __shfl_xor(s, msk, 32);
    if (lane == 0) ap[m * H_ + h] = s;
}

// decay[b,t,h] = exp(-exp(A_log[h]) * softplus(ap + dt_bias[h]))
__global__ void decay_kernel(const float* __restrict__ ap,
                             const float* __restrict__ A_log,
                             const float* __restrict__ dt_bias,
                             float* __restrict__ dec) {
    const int idx = blockIdx.x * blockDim.x + threadIdx.x;
    if (idx >= MROWS * H_) return;
    const int h = idx % H_;
    float z  = ap[idx] + dt_bias[h];
    float sp = (z > 20.0f) ? z : log1pf(__expf(z));
    dec[idx] = __expf(-__expf(A_log[h]) * sp);
}

// Depthwise causal conv (K=4, left pad 3) + SiLU. pre/out: [B,T,C], w: [C,1,DC]
__global__ void dwconv_silu(const float* __restrict__ pre, const float* __restrict__ w,
                            float* __restrict__ out, int C) {
    const int idx = blockIdx.x * blockDim.x + threadIdx.x;
    if (idx >= B_ * T_ * C) return;
    const int c = idx % C;
    const int t = (idx / C) % T_;
    const int b = idx / (C * T_);
    const float* wc = w + c * DC_;
    float acc = 0.0f;
#pragma unroll
    for (int j = 0; j < DC_; ++j) {
        const int ts = t + j - (DC_ - 1);
        if (ts >= 0) acc = fmaf(pre[((size_t)b * T_ + ts) * C + c], wc[j], acc);
    }
    out[idx] = fast_silu(acc);
}

// ---------------------------------------------------------------------------
// Sequential scan. One block per (b,h), 64 threads (2 waves).
// Lane `col` (0..63) owns column `col` of the 32x64 state in 32 VGPRs.
// wave0 normalizes k_t, wave1 normalizes q_t (butterfly) -> double-buffered
// LDS broadcast arrays; ONE barrier per step (step t+1's barrier orders step-t
// reads before step-t+2 writes to the same buffer). Next step's k/q/v/decay
// are prefetched before the barrier to hide global latency.
// ---------------------------------------------------------------------------
__global__ __launch_bounds__(64)
void e88_scan(const float* __restrict__ q, const float* __restrict__ k,
              const float* __restrict__ v, const float* __restrict__ decay,
              float* __restrict__ attn_out, float* __restrict__ S_final) {
    const int bh   = blockIdx.x;      // 0..B*H-1
    const int b    = bh / H_;
    const int h    = bh % H_;
    const int tid  = threadIdx.x;     // 0..63
    const int wave = tid >> 5;
    const int lane = tid & 31;
    const int col  = tid;             // owned HV column

    __shared__ float kn_s[2][N_];
    __shared__ float qn_s[2][N_];

    const float* kq_p = ((wave == 0) ? k : q)
                        + (size_t)(b * T_) * (H_ * N_) + h * N_ + lane;
    const float* v_p  = v     + (size_t)(b * T_) * (H_ * HV_) + h * HV_ + col;
    const float* d_p  = decay + (size_t)(b * T_) * H_ + h;
    float*       o_p  = attn_out + (size_t)(b * T_) * (H_ * HV_) + h * HV_ + col;

    float S[N_];
#pragma unroll
    for (int n = 0; n < N_; ++n) S[n] = 0.0f;

    // preload step 0
    float kq = kq_p[0];
    float vv = v_p[0];
    float dd = d_p[0];

    for (int t = 0; t < T_; ++t) {
        const int buf = t & 1;
        {   // normalize k (wave0) / q (wave1)
            float ss = kq * kq;
#pragma unroll
            for (int msk = 16; msk >= 1; msk >>= 1) ss += __shfl_xor(ss, msk, 32);
            const float nv = kq * (1.0f / (sqrtf(ss) + 1e-6f));
            if (wave == 0) kn_s[buf][lane] = nv; else qn_s[buf][lane] = nv;
        }

        // prefetch step t+1 (uniform branch) before the barrier
        float kq_n = 0.0f, vv_n = 0.0f, dd_n = 0.0f;
        if (t + 1 < T_) {
            kq_n = kq_p[(size_t)(t + 1) * (H_ * N_)];
            vv_n = v_p[(size_t)(t + 1) * (H_ * HV_)];
            dd_n = d_p[(size_t)(t + 1) * H_];
        }

        __syncthreads();

        float kc[N_], qc[N_];
        float r = 0.0f;
#pragma unroll
        for (int n = 0; n < N_; ++n) {
            kc[n] = kn_s[buf][n];
            qc[n] = qn_s[buf][n];
            r = fmaf(S[n], kc[n], r);
        }
        const float delta = vv - r;

        float sq = 0.0f;
#pragma unroll
        for (int n = 0; n < N_; ++n) {
            const float z  = fmaf(dd, S[n], kc[n] * delta);
            const float sn = fast_tanh(z);
            S[n] = sn;
            sq = fmaf(sn, qc[n], sq);
        }
        const float o = sq * sq * fast_sigmoid(sq);
        o_p[(size_t)t * (H_ * HV_)] = o;

        kq = kq_n; vv = vv_n; dd = dd_n;
    }

#pragma unroll
    for (int n = 0; n < N_; ++n)
        S_final[((size_t)(b * H_ + h) * N_ + n) * HV_ + col] = S[n];
}

// RMS-norm over HV + gated sigmoid. One block per (b,t,h), 64 threads.
__global__ __launch_bounds__(64)
void rms_gate(const float* __restrict__ attn, const float* __restrict__ graw,
              const float* __restrict__ onw, float* __restrict__ gated) {
    const int g    = blockIdx.x;       // (b*T+t)*H + h
    const int i    = threadIdx.x;      // 0..63
    const int wave = i >> 5;
    __shared__ float ss2[2];

    const size_t idx = (size_t)g * HV_ + i;   // == (b*T+t)*VD + h*HV + i
    const float o = attn[idx];
    float p = o * o;
#pragma unroll
    for (int msk = 16; msk >= 1; msk >>= 1) p += __shfl_xor(p, msk, 32);
    if ((i & 31) == 0) ss2[wave] = p;
    __syncthreads();
    const float mean = (ss2[0] + ss2[1]) * (1.0f / HV_);
    const float rms  = rsqrtf(mean + 1e-5f);
    gated[idx] = o * rms * onw[i] * fast_sigmoid(graw[idx]);
}

// ---------------------------------------------------------------------------
extern "C" void kernel_launch(void* const* d_in, const int* in_sizes, int n_in,
                              void* d_out, int out_size, void* d_ws, size_t ws_size,
                              hipStream_t stream) {
    const float* x       = (const float*)d_in[0];
    const float* Wq      = (const float*)d_in[1];
    const float* Wk      = (const float*)d_in[2];
    const float* Wv      = (const float*)d_in[3];
    const float* Wa      = (const float*)d_in[4];
    const float* A_log   = (const float*)d_in[5];
    const float* dt_bias = (const float*)d_in[6];
    const float* qcw     = (const float*)d_in[7];
    const float* kcw     = (const float*)d_in[8];
    const float* vcw     = (const float*)d_in[9];
    const float* Wg      = (const float*)d_in[10];
    const float* Wo      = (const float*)d_in[11];
    const float* onw     = (const float*)d_in[12];

    float* ws = (float*)d_ws;
    const size_t M = MROWS;
    float* q_pre  = ws;                       // 1M floats
    float* k_pre  = q_pre  + M * KD_;         // 1M
    float* v_pre  = k_pre  + M * KD_;         // 2M
    float* g_raw  = v_pre  + M * VD_;         // 2M
    float* q_post = g_raw  + M * VD_;         // 1M
    float* k_post = q_post + M * KD_;         // 1M
    float* v_post = k_post + M * KD_;         // 2M
    float* ap     = v_post + M * VD_;         // 32K
    float* dec    = ap     + M * H_;          // 32K
    float* attn   = q_pre;                    // reuse q_pre+k_pre (2M, contiguous)
    float* gated  = v_pre;                    // reuse v_pre (2M)

    float* y       = (float*)d_out;
    float* S_final = y + (size_t)B_ * T_ * DIM;

    // Input projections (WMMA f32, 16x128 tile per wave, pipelined K loop)
    gemm_xwT_wmma<<<dim3(KD_ / 128, M / 16), 32, 0, stream>>>(x, Wq, q_pre, DIM, KD_);
    gemm_xwT_wmma<<<dim3(KD_ / 128, M / 16), 32, 0, stream>>>(x, Wk, k_pre, DIM, KD_);
    gemm_xwT_wmma<<<dim3(VD_ / 128, M / 16), 32, 0, stream>>>(x, Wv, v_pre, DIM, VD_);
    gemm_xwT_wmma<<<dim3(VD_ / 128, M / 16), 32, 0, stream>>>(x, Wg, g_raw, DIM, VD_);
    a_proj_kernel<<<(M * H_) / 8, 256, 0, stream>>>(x, Wa, ap);
    decay_kernel<<<(M * H_ + 255) / 256, 256, 0, stream>>>(ap, A_log, dt_bias, dec);

    // Depthwise conv + SiLU
    dwconv_silu<<<(M * KD_ + 255) / 256, 256, 0, stream>>>(q_pre, qcw, q_post, KD_);
    dwconv_silu<<<(M * KD_ + 255) / 256, 256, 0, stream>>>(k_pre, kcw, k_post, KD_);
    dwconv_silu<<<(M * VD_ + 255) / 256, 256, 0, stream>>>(v_pre, vcw, v_post, VD_);

    // Sequential recurrence (latency-bound; register-resident state)
    e88_scan<<<B_ * H_, 64, 0, stream>>>(q_post, k_post, v_post, dec, attn, S_final);

    // RMS-norm + gate, then output projection (WMMA f32)
    rms_gate<<<B_ * T_ * H_, 64, 0, stream>>>(attn, g_raw, onw, gated);
    gemm_xwT_wmma<<<dim3(DIM / 128, M / 16), 32, 0, stream>>>(gated, Wo, y, VD_, DIM);
}